// GPLightEncoder_44702019617436
// MI455X (gfx1250) — compile-verified
//
#include <hip/hip_runtime.h>
#include <math.h>

// ---------------------------------------------------------------------------
// GNN encoder for MI455X (gfx1250, wave32).
// h0 = x@W_in + b_in
// per layer: agg = segment_sum(h[src], dst) ; h = elu(h@W_self + b + (agg/cnt)@W_nbr)
// (reassociated: segment_sum(h[src]@W) == segment_sum(h[src])@W)
// Matmuls use V_WMMA_F32_16X16X4_F32 tiles.
// ---------------------------------------------------------------------------

#define N_NODES 50000
#define N_EDGES 800000
#define IN_FEAT 64
#define HIDDEN  128
#define NUM_LAYERS 3

typedef float v2f __attribute__((ext_vector_type(2)));
typedef float v8f __attribute__((ext_vector_type(8)));

// ---------------- utility ----------------
__global__ void zero_f32_kernel(float* __restrict__ p, int n) {
    int i = blockIdx.x * blockDim.x + threadIdx.x;
    if (i < n) p[i] = 0.0f;
}

__global__ void count_edges_kernel(const int* __restrict__ dst,
                                   float* __restrict__ cnt, int E) {
    int e = blockIdx.x * blockDim.x + threadIdx.x;
    if (e < E) atomicAdd(&cnt[dst[e]], 1.0f);
}

// ---------------- input projection: h = x @ W_in + b_in -------------------
// grid: (N/16, HIDDEN/(16*4)), block: 128 (4 waves). One wave = one 16x16 tile.
__global__ __launch_bounds__(128)
void input_proj_kernel(const float* __restrict__ x,    // [N, 64]
                       const float* __restrict__ Win,  // [64, 128]
                       const float* __restrict__ bin,  // [128]
                       float* __restrict__ h)          // [N, 128]
{
    const int lane = threadIdx.x & 31;
    const int wave = threadIdx.x >> 5;
    const int half = lane >> 4;      // K sub-group selector
    const int r    = lane & 15;      // A-row / B-col / D-col index
    const int mBase = blockIdx.x * 16;
    const int nBase = (blockIdx.y * 4 + wave) * 16;

    const float* xrow = x + (size_t)(mBase + r) * IN_FEAT;

    v8f acc = {};
#pragma unroll
    for (int k0 = 0; k0 < IN_FEAT; k0 += 4) {
        const int ka = k0 + 2 * half;
        v2f a = *(const v2f*)(xrow + ka);                 // A[r][ka..ka+1]
        v2f b;
        b.x = Win[(size_t)(ka + 0) * HIDDEN + nBase + r]; // B[ka][n]
        b.y = Win[(size_t)(ka + 1) * HIDDEN + nBase + r]; // B[ka+1][n]
        acc = __builtin_amdgcn_wmma_f32_16x16x4_f32(false, a, false, b,
                                                    (short)0, acc, false, false);
    }
    const float bias = bin[nBase + r];
#pragma unroll
    for (int v = 0; v < 8; ++v) {
        const int row = v + 8 * half;
        h[(size_t)(mBase + row) * HIDDEN + nBase + r] = acc[v] + bias;
    }
}

// ---------------- edge scatter: agg[dst] += h[src] ------------------------
// One wave per edge; each lane moves one float4 (32 lanes * 16B = 512B = row).
__global__ __launch_bounds__(256)
void scatter_kernel(const int* __restrict__ src, const int* __restrict__ dst,
                    const float* __restrict__ h, float* __restrict__ agg, int E)
{
    const int wid  = (blockIdx.x * blockDim.x + threadIdx.x) >> 5;
    const int lane = threadIdx.x & 31;
    if (wid >= E) return;
    const int s = src[wid];
    const int d = dst[wid];
    const float4 v = *(const float4*)(h + (size_t)s * HIDDEN + lane * 4);
    float* o = agg + (size_t)d * HIDDEN + lane * 4;
    atomicAdd(o + 0, v.x);
    atomicAdd(o + 1, v.y);
    atomicAdd(o + 2, v.z);
    atomicAdd(o + 3, v.w);
}

// ---------------- layer: hout = elu(hin@Ws + b + (agg/cnt)@Wn) ------------
__global__ __launch_bounds__(128)
void layer_kernel(const float* __restrict__ hin,   // [N,128]
                  const float* __restrict__ agg,   // [N,128]
                  const float* __restrict__ cnt,   // [N]
                  const float* __restrict__ Ws,    // [128,128]
                  const float* __restrict__ bs,    // [128]
                  const float* __restrict__ Wn,    // [128,128]
                  float* __restrict__ hout)        // [N,128]
{
    const int lane = threadIdx.x & 31;
    const int wave = threadIdx.x >> 5;
    const int half = lane >> 4;
    const int r    = lane & 15;
    const int mBase = blockIdx.x * 16;
    const int nBase = (blockIdx.y * 4 + wave) * 16;

    const int arow = mBase + r;                       // A-tile row this lane feeds
    const float inv = 1.0f / fmaxf(cnt[arow], 1.0f);  // scatter-mean denom
    const float* hrow = hin + (size_t)arow * HIDDEN;
    const float* grow = agg + (size_t)arow * HIDDEN;

    v8f acc = {};
#pragma unroll 8
    for (int k0 = 0; k0 < HIDDEN; k0 += 4) {
        const int ka = k0 + 2 * half;
        // self term: hin @ Ws
        v2f a = *(const v2f*)(hrow + ka);
        v2f b;
        b.x = Ws[(size_t)(ka + 0) * HIDDEN + nBase + r];
        b.y = Ws[(size_t)(ka + 1) * HIDDEN + nBase + r];
        acc = __builtin_amdgcn_wmma_f32_16x16x4_f32(false, a, false, b,
                                                    (short)0, acc, false, false);
        // neighbor term: (agg/cnt) @ Wn  (row scale folded into A load)
        v2f a2 = *(const v2f*)(grow + ka);
        a2.x *= inv; a2.y *= inv;
        v2f b2;
        b2.x = Wn[(size_t)(ka + 0) * HIDDEN + nBase + r];
        b2.y = Wn[(size_t)(ka + 1) * HIDDEN + nBase + r];
        acc = __builtin_amdgcn_wmma_f32_16x16x4_f32(false, a2, false, b2,
                                                    (short)0, acc, false, false);
    }
    const float bias = bs[nBase + r];
#pragma unroll
    for (int v = 0; v < 8; ++v) {
        const int row = v + 8 * half;
        float val = acc[v] + bias;
        val = (val > 0.0f) ? val : expm1f(val);       // ELU, alpha=1
        hout[(size_t)(mBase + row) * HIDDEN + nBase + r] = val;
    }
}

// ---------------------------------------------------------------------------
extern "C" void kernel_launch(void* const* d_in, const int* in_sizes, int n_in,
                              void* d_out, int out_size, void* d_ws, size_t ws_size,
                              hipStream_t stream) {
    const float* x     = (const float*)d_in[0];          // [N, 64]
    const int*   ei    = (const int*)  d_in[1];          // [2, E]
    const float* Win   = (const float*)d_in[2];          // [64, 128]
    const float* bin   = (const float*)d_in[3];          // [128]
    const float* Wself = (const float*)d_in[4];          // [3,128,128]
    const float* bself = (const float*)d_in[5];          // [3,128]
    const float* Wnbr  = (const float*)d_in[6];          // [3,128,128]
    float*       out   = (float*)d_out;                  // [N, 128]

    const int* src = ei;
    const int* dst = ei + N_EDGES;

    // workspace layout: hA [N*H] | agg [N*H] | cnt [N]   (~51.4 MB)
    float* hA  = (float*)d_ws;
    float* agg = hA  + (size_t)N_NODES * HIDDEN;
    float* cnt = agg + (size_t)N_NODES * HIDDEN;

    const int NH = N_NODES * HIDDEN;

    // edge counts (grad-constant, compute once)
    zero_f32_kernel<<<(N_NODES + 255) / 256, 256, 0, stream>>>(cnt, N_NODES);
    count_edges_kernel<<<(N_EDGES + 255) / 256, 256, 0, stream>>>(dst, cnt, N_EDGES);

    // h0 = x @ W_in + b_in  -> hA
    dim3 gIn(N_NODES / 16, HIDDEN / 64);   // 3125 x 2, 4 waves/block
    input_proj_kernel<<<gIn, 128, 0, stream>>>(x, Win, bin, hA);

    // layer ping-pong: hA -> out -> hA -> out
    const float* hcur = hA;
    float* bufs[NUM_LAYERS] = { out, hA, out };
    const int scatterBlocks = (N_EDGES * 32 + 255) / 256;
    for (int l = 0; l < NUM_LAYERS; ++l) {
        zero_f32_kernel<<<(NH + 255) / 256, 256, 0, stream>>>(agg, NH);
        scatter_kernel<<<scatterBlocks, 256, 0, stream>>>(src, dst, hcur, agg, N_EDGES);
        layer_kernel<<<gIn, 128, 0, stream>>>(hcur, agg, cnt,
                                              Wself + (size_t)l * HIDDEN * HIDDEN,
                                              bself + (size_t)l * HIDDEN,
                                              Wnbr  + (size_t)l * HIDDEN * HIDDEN,
                                              bufs[l]);
        hcur = bufs[l];
    }
}